// SoftmaxRandomSamplePolicy_8194797600967
// MI455X (gfx1250) — compile-verified
//
#include <hip/hip_runtime.h>
#include <stdint.h>

// Problem shape (fixed by the reference)
#define BATCH    4096
#define VOCAB    50257
#define NT       256               // threads per block (8 wave32 waves)
#define TILE     2048              // elements per tile per array
#define NTILES   (VOCAB / TILE)    // 24 full tiles over the aligned region
#define STAGES   4                 // LDS pipeline depth (3 tiles in flight)
#define L2E      1.4426950408889634f
#define NEGINF   (-3.0e30f)

// ---- CDNA5 async global->LDS copy path (gfx1250) --------------------------
// Generic pointers to __shared__ carry the wave-relative LDS byte offset in
// their low 32 bits, which is what the async-load VDST operand wants.
__device__ __forceinline__ uint32_t lds_addr(const void* p) {
    return (uint32_t)(uintptr_t)p;
}

// GVS form: global addr = SGPR64 base + VGPR32 offset + INST_OFFSET.
// Per the ISA, INST_OFFSET is ALSO added to the LDS destination address, so
// the LDS layout below keeps chunk stride == global chunk stride (4096 B).
__device__ __forceinline__ void async_ld_b128(uint32_t dst_lds, uint32_t voff,
                                              const float* sbase, int imm) {
    asm volatile("global_load_async_to_lds_b128 %0, %1, %2 offset:%3"
                 :: "v"(dst_lds), "v"(voff), "s"(sbase), "n"(imm)
                 : "memory");
}

template <int N>
__device__ __forceinline__ void wait_asynccnt() {
    asm volatile("s_wait_asynccnt %0" :: "n"(N) : "memory");
}
// ---------------------------------------------------------------------------

__global__ __launch_bounds__(NT)
void SoftmaxRandomSamplePolicy_8194797600967_kernel(const float* __restrict__ logits,
                                                    const float* __restrict__ gumbel,
                                                    float* __restrict__ out) {
    // [stage][logits|gumbel][elem] : 4*2*2048*4 = 64 KB (of 320 KB/WGP)
    __shared__ __align__(16) float smem[STAGES][2][TILE];

    const int t   = threadIdx.x;
    const int row = blockIdx.x;
    const float* __restrict__ Lrow = logits + (size_t)row * VOCAB;
    const float* __restrict__ Grow = gumbel + (size_t)row * VOCAB;

    // Row byte base = row*50257*4 ≡ 4*row (mod 16): peel h = (-row) mod 4
    // elements so the bulk region is 16B aligned for b128 asyncs.
    const int h = (4 - (row & 3)) & 3;

    // Per-thread state: online logsumexp (m,s) + argmax of key = x + g
    float m = NEGINF, s = 0.0f, bk = NEGINF;
    int   bi = VOCAB;

    // ---- alignment head: elements [0, h), h in 0..3 ----
    if (t < h) {
        const float xv = Lrow[t], gv = Grow[t];
        const float key = xv + gv;
        if (key > bk) { bk = key; bi = t; }
        m = xv;
        s = 1.0f;                   // exp(xv - xv)
    }

    // Thread t owns global elements  h + i*2048 + c*1024 + t*4 + j  (j=0..3)
    // and the matching LDS bytes     st*16K + a*8K + c*4096 + t*16.
    const uint32_t vbase = (uint32_t)(h * 4 + t * 16);

    auto issue = [&](int tile) {
        const int      st = tile & (STAGES - 1);
        const uint32_t vo = vbase + (uint32_t)tile * (TILE * 4);
        const uint32_t dL = lds_addr(&smem[st][0][t * 4]);
        const uint32_t dG = lds_addr(&smem[st][1][t * 4]);
        async_ld_b128(dL, vo, Lrow, 0);      // chunk 0, logits
        async_ld_b128(dL, vo, Lrow, 4096);   // chunk 1, logits
        async_ld_b128(dG, vo, Grow, 0);      // chunk 0, gumbel
        async_ld_b128(dG, vo, Grow, 4096);   // chunk 1, gumbel
    };

    auto consume = [&](int tile) {
        const int st = tile & (STAGES - 1);
        const int tb = h + tile * TILE + t * 4;

        const float4 xa = *(const float4*)&smem[st][0][t * 4];
        const float4 xb = *(const float4*)&smem[st][0][1024 + t * 4];
        const float4 ga = *(const float4*)&smem[st][1][t * 4];
        const float4 gb = *(const float4*)&smem[st][1][1024 + t * 4];
        const float x[8] = {xa.x, xa.y, xa.z, xa.w, xb.x, xb.y, xb.z, xb.w};
        const float g[8] = {ga.x, ga.y, ga.z, ga.w, gb.x, gb.y, gb.z, gb.w};

        // argmax of (logit + gumbel); ascending idx + strict '>' keeps the
        // first index on (measure-zero) ties
#pragma unroll
        for (int j = 0; j < 8; ++j) {
            const float key = x[j] + g[j];
            const int   id  = tb + (j & 3) + ((j >> 2) << 10);
            if (key > bk) { bk = key; bi = id; }
        }

        // online logsumexp, exp2 domain: 9 trans + 9 FMA per 8 elements
        float xm = fmaxf(fmaxf(fmaxf(x[0], x[1]), fmaxf(x[2], x[3])),
                         fmaxf(fmaxf(x[4], x[5]), fmaxf(x[6], x[7])));
        const float nm = fmaxf(m, xm);
        const float nl = nm * L2E;
        float acc = s * exp2f(fmaf(m, L2E, -nl));
#pragma unroll
        for (int j = 0; j < 8; ++j) acc += exp2f(fmaf(x[j], L2E, -nl));
        s = acc;
        m = nm;
    };

    // ---- software pipeline: 3 tiles in flight (12 outstanding asyncs) ----
    issue(0); issue(1); issue(2);
#pragma unroll 1
    for (int i = 0; i < NTILES - 3; ++i) {
        issue(i + 3);
        wait_asynccnt<12>();
        consume(i);
    }
    wait_asynccnt<8>();  consume(NTILES - 3);
    wait_asynccnt<4>();  consume(NTILES - 2);
    wait_asynccnt<0>();  consume(NTILES - 1);

    // ---- tail: elements [h + 49152, 50257), 1102..1105 of them ----
    const int tstart = h + NTILES * TILE;
#pragma unroll
    for (int k = 0; k < 5; ++k) {
        const int idx = tstart + k * NT + t;
        if (idx < VOCAB) {
            const float xv = Lrow[idx], gv = Grow[idx];
            const float key = xv + gv;
            if (key > bk) { bk = key; bi = idx; }
            const float nm = fmaxf(m, xv);
            s = s * __expf(m - nm) + __expf(xv - nm);
            m = nm;
        }
    }

    // ---- wave32 reduction ----
#pragma unroll
    for (int off = 16; off > 0; off >>= 1) {
        const float om = __shfl_xor(m,  off, 32);
        const float os = __shfl_xor(s,  off, 32);
        const float ok = __shfl_xor(bk, off, 32);
        const int   oi = __shfl_xor(bi, off, 32);
        const float nm = fmaxf(m, om);
        s = s * __expf(m - nm) + os * __expf(om - nm);
        m = nm;
        if (ok > bk || (ok == bk && oi < bi)) { bk = ok; bi = oi; }
    }

    // ---- cross-wave reduction via reused LDS (all asyncs complete) ----
    const int lane = t & 31;
    const int w    = t >> 5;
    float* red  = &smem[0][0][0];        // 8 waves * 3 floats
    int*   redi = (int*)&smem[0][1][0];  // 8 ints
    __syncthreads();
    if (lane == 0) {
        red[w]      = m;
        red[8 + w]  = s;
        red[16 + w] = bk;
        redi[w]     = bi;
    }
    __syncthreads();
    if (t == 0) {
        float M = red[0], S = red[8], K = red[16];
        int   I = redi[0];
#pragma unroll
        for (int w2 = 1; w2 < 8; ++w2) {
            const float om = red[w2], os = red[8 + w2], ok = red[16 + w2];
            const int   oi = redi[w2];
            const float nm = fmaxf(M, om);
            S = S * __expf(M - nm) + os * __expf(om - nm);
            M = nm;
            if (ok > K || (ok == K && oi < I)) { K = ok; I = oi; }
        }
        out[row]         = (float)I;                  // idx < 2^24: exact
        out[BATCH + row] = Lrow[I] - (M + __logf(S)); // log_softmax at idx
    }
}

extern "C" void kernel_launch(void* const* d_in, const int* in_sizes, int n_in,
                              void* d_out, int out_size, void* d_ws, size_t ws_size,
                              hipStream_t stream) {
    (void)in_sizes; (void)n_in; (void)out_size; (void)d_ws; (void)ws_size;
    const float* logits = (const float*)d_in[0];
    const float* gumbel = (const float*)d_in[1];
    float*       out    = (float*)d_out;
    SoftmaxRandomSamplePolicy_8194797600967_kernel<<<BATCH, NT, 0, stream>>>(logits, gumbel, out);
}